// TopK_21303037788693
// MI455X (gfx1250) — compile-verified
//
#include <hip/hip_runtime.h>
#include <math.h>

// ---------------------------------------------------------------------------
// TopK rank-1 projection + gather, MI455X (gfx1250, wave32)
//
// Stage 1: scores[i] = dot(embs[i,:], w) / ||w||  -- V_WMMA_F32_16X16X4_F32,
//          branch-free B build, fully unrolled K loop (32 WMMAs / wave-tile).
//          Memory bound: 512MB @ 23.3TB/s ~ 22us floor.
// Stage 2: 4096-bin histogram -> threshold bin -> deterministic compaction
//          (ballot prefix sums; no order-nondeterministic atomics).
// Stage 3: gather rows via GLOBAL_LOAD_ASYNC_TO_LDS_B128 (ASYNCcnt path),
//          tanh gate applied on the transposed write-out.
// ---------------------------------------------------------------------------

#define NFEATS 128
#define KSEL   50000
#define NBINS  4096
#define CHUNK  8192            // elements per block in count/write passes
#define TSTRIDE 132            // LDS tile stride (floats): 132*4 = 528 = 16*33

typedef __attribute__((ext_vector_type(2))) float v2f;
typedef __attribute__((ext_vector_type(8))) float v8f;
typedef __attribute__((ext_vector_type(4))) int   v4i;

#if defined(__gfx1250__) && __has_builtin(__builtin_amdgcn_global_load_async_to_lds_b128)
#define HAVE_ASYNC_LDS 1
// builtin signature (from hipcc diagnostic): (int4 as(1)*, int4 as(3)*, Ii, Ii)
typedef __attribute__((address_space(1))) v4i* gp_v4i;
typedef __attribute__((address_space(3))) v4i* lp_v4i;
#endif

__device__ __forceinline__ void wait_asynccnt0() {
#if __has_builtin(__builtin_amdgcn_s_wait_asynccnt)
    __builtin_amdgcn_s_wait_asynccnt(0);
#else
    asm volatile("s_wait_asynccnt 0x0" ::: "memory");
#endif
}

__device__ __forceinline__ unsigned ordkey(float f) {
    // order-preserving map: larger float -> larger unsigned
    unsigned u = __float_as_uint(f);
    return (u & 0x80000000u) ? ~u : (u | 0x80000000u);
}

// ---------------------------------------------------------------------------
// Stage 1: scores via WMMA f32 16x16x4. One wave -> 16 rows (n % 16 == 0).
// A tile (16x4 f32): lane L holds row M=L%16, K pair {0,1} (L<16) or {2,3}.
// B tile (4x16 f32): column 0 = scorer chunk; built branch-free with a 0/1
//   selector so EXEC stays all-ones and no saveexec appears in the loop.
// D column 0: lane 0 -> M=0..7 in c[0..7], lane 16 -> M=8..15.
// ---------------------------------------------------------------------------
__global__ void topk_scores_wmma(const float* __restrict__ embs,
                                 const float* __restrict__ w,
                                 float* __restrict__ scores, int n) {
    const int wave = threadIdx.x >> 5;
    const int lane = threadIdx.x & 31;
    const int base = (blockIdx.x * 8 + wave) * 16;
    if (base >= n) return;                 // wave-uniform: EXEC stays all-ones

    // 1/||w|| (uniform, constant-cached; negligible vs the 512MB stream)
    float ss = 0.f;
#pragma unroll
    for (int i = 0; i < NFEATS; ++i) { float x = w[i]; ss += x * x; }
    const float inv_norm = 1.0f / fmaxf(sqrtf(ss), 1e-6f);

    const int   r    = base + (lane & 15);           // always < n (n%16==0)
    const int   koff = (lane < 16) ? 0 : 2;
    const float sel  = ((lane & 15) == 0) ? 1.0f : 0.0f;   // lanes 0 and 16
    const float* rowp = embs + (size_t)r * NFEATS;
    const float* wlp  = w + koff;

    v8f c = {};
#pragma unroll
    for (int kk = 0; kk < NFEATS; kk += 4) {
        v2f a = *(const v2f*)(rowp + kk + koff);     // global_load_b64
        v2f wv = *(const v2f*)(wlp + kk);            // cached, per lane-group
        v2f b; b.x = sel * wv.x; b.y = sel * wv.y;   // branch-free column 0
        c = __builtin_amdgcn_wmma_f32_16x16x4_f32(
                false, a, false, b, (short)0, c, false, false);
    }

    if (lane == 0 || lane == 16) {
        const int m0 = (lane == 16) ? 8 : 0;
#pragma unroll
        for (int i = 0; i < 8; ++i)
            scores[base + m0 + i] = c[i] * inv_norm;
    }
}

// ---------------------------------------------------------------------------
// Zero the selection scratch (hist, meta, output index/value buffers).
// ---------------------------------------------------------------------------
__global__ void topk_init(unsigned* hist, unsigned* meta,
                          int* idxb, float* valb) {
    int t = blockIdx.x * blockDim.x + threadIdx.x;
    if (t < NBINS) hist[t] = 0u;
    if (t < 16)    meta[t] = 0u;
    if (t < KSEL)  { idxb[t] = 0; valb[t] = 0.f; }
}

// ---------------------------------------------------------------------------
// Stage 2a: 4096-bin histogram of score keys (LDS-privatized).
// ---------------------------------------------------------------------------
__global__ void topk_hist(const float* __restrict__ scores,
                          unsigned* __restrict__ hist, int n) {
    __shared__ unsigned sh[NBINS];
    for (int i = threadIdx.x; i < NBINS; i += blockDim.x) sh[i] = 0u;
    __syncthreads();
    const int stride = gridDim.x * blockDim.x;
    for (int i = blockIdx.x * blockDim.x + threadIdx.x; i < n; i += stride) {
        __builtin_prefetch(scores + i + stride, 0, 0);
        atomicAdd(&sh[ordkey(scores[i]) >> 20], 1u);
    }
    __syncthreads();
    for (int i = threadIdx.x; i < NBINS; i += blockDim.x) {
        unsigned v = sh[i];
        if (v) atomicAdd(&hist[i], v);
    }
}

// ---------------------------------------------------------------------------
// Stage 2b: walk histogram from the top, find threshold bin.
// ---------------------------------------------------------------------------
__global__ void topk_thresh(const unsigned* __restrict__ hist,
                            unsigned* __restrict__ meta) {
    if (threadIdx.x == 0) {
        unsigned cum = 0; int tb = 0;
        for (int b = NBINS - 1; b >= 0; --b) {
            unsigned c = hist[b];
            if (cum + c >= (unsigned)KSEL) { tb = b; break; }
            cum += c;
        }
        meta[0] = (unsigned)tb;   // threshold bin
        meta[1] = cum;            // strictly-above count (by bin)
    }
}

// ---------------------------------------------------------------------------
// Stage 2c: per-block counts of (bin>tb, bin==tb) over contiguous chunks.
// ---------------------------------------------------------------------------
__global__ void topk_count(const float* __restrict__ scores,
                           const unsigned* __restrict__ meta,
                           unsigned* __restrict__ bc_gt,
                           unsigned* __restrict__ bc_eq, int n) {
    __shared__ unsigned s_gt, s_eq;
    if (threadIdx.x == 0) { s_gt = 0u; s_eq = 0u; }
    __syncthreads();
    const unsigned tb = meta[0];
    const int base = blockIdx.x * CHUNK;
    unsigned lgt = 0, leq = 0;
    for (int off = threadIdx.x; off < CHUNK; off += blockDim.x) {
        int i = base + off;
        if (i < n) {
            unsigned b = ordkey(scores[i]) >> 20;
            lgt += (b > tb); leq += (b == tb);
        }
    }
    atomicAdd(&s_gt, lgt); atomicAdd(&s_eq, leq);
    __syncthreads();
    if (threadIdx.x == 0) { bc_gt[blockIdx.x] = s_gt; bc_eq[blockIdx.x] = s_eq; }
}

// ---------------------------------------------------------------------------
// Stage 2d: exclusive scan over block counts (tiny, serial).
// ---------------------------------------------------------------------------
__global__ void topk_scan(unsigned* bc_gt, unsigned* bc_eq,
                          unsigned* meta, int nblocks) {
    if (threadIdx.x == 0) {
        unsigned rg = 0, re = 0;
        for (int i = 0; i < nblocks; ++i) {
            unsigned g = bc_gt[i], e = bc_eq[i];
            bc_gt[i] = rg; bc_eq[i] = re;
            rg += g; re += e;
        }
        meta[3] = rg;                               // total strictly-above
        meta[4] = (rg < KSEL) ? (KSEL - rg) : 0u;   // fill from threshold bin
    }
}

// ---------------------------------------------------------------------------
// Stage 2e: deterministic compaction. Rank = index order via wave32 ballot
// prefix + cross-wave LDS scan.
// ---------------------------------------------------------------------------
__global__ void topk_write(const float* __restrict__ scores,
                           const unsigned* __restrict__ meta,
                           const unsigned* __restrict__ bc_gt,
                           const unsigned* __restrict__ bc_eq,
                           int* __restrict__ idx_out,
                           float* __restrict__ val_out, int n) {
    __shared__ unsigned s_wg[8], s_we[8];
    __shared__ unsigned s_rg, s_re;
    const unsigned tb       = meta[0];
    const unsigned total_gt = meta[3];
    const unsigned need_eq  = meta[4];
    if (threadIdx.x == 0) { s_rg = bc_gt[blockIdx.x]; s_re = bc_eq[blockIdx.x]; }
    __syncthreads();
    const int base = blockIdx.x * CHUNK;
    const int wave = threadIdx.x >> 5;
    const int lane = threadIdx.x & 31;
    for (int round = 0; round < CHUNK / 256; ++round) {
        int i = base + round * 256 + threadIdx.x;
        bool gt = false, eq = false;
        float v = 0.f;
        if (i < n) {
            v = scores[i];
            unsigned b = ordkey(v) >> 20;
            gt = (b > tb); eq = (b == tb);
        }
        unsigned long long mg = __ballot(gt);
        unsigned long long me = __ballot(eq);
        unsigned long long lower = (1ull << lane) - 1ull;
        unsigned pg = (unsigned)__popcll(mg & lower);
        unsigned pe = (unsigned)__popcll(me & lower);
        if (lane == 0) { s_wg[wave] = (unsigned)__popcll(mg);
                         s_we[wave] = (unsigned)__popcll(me); }
        __syncthreads();
        unsigned og = s_rg, oe = s_re;
        for (int wv = 0; wv < wave; ++wv) { og += s_wg[wv]; oe += s_we[wv]; }
        if (gt) {
            unsigned pos = og + pg;
            idx_out[pos] = i; val_out[pos] = v;
        }
        if (eq) {
            unsigned p = oe + pe;
            if (p < need_eq) { idx_out[total_gt + p] = i;
                               val_out[total_gt + p] = v; }
        }
        __syncthreads();
        if (threadIdx.x == 0) {
            for (int wv = 0; wv < 8; ++wv) { s_rg += s_wg[wv]; s_re += s_we[wv]; }
        }
        __syncthreads();
    }
}

// ---------------------------------------------------------------------------
// Stage 3: gather rows raw into LDS (async global->LDS copies, ASYNCcnt),
// then gate by tanh(score) while writing the transposed [FEATS, K] output.
// 64 columns per block; LDS tile 64 x TSTRIDE floats (16B-aligned rows).
// ---------------------------------------------------------------------------
__global__ void topk_gather(const float* __restrict__ embs,
                            const int* __restrict__ idx,
                            const float* __restrict__ vals,
                            float* __restrict__ out, int n) {
    __shared__ float tile[64 * TSTRIDE];
    const int j0  = blockIdx.x * 64;
    const int tid = threadIdx.x;           // 256 threads

    // Phase 1: copy 64 rows x 128 floats, 16B per lane per op.
    // e in [0, 2048): lr = e/32 (row), q = e%32 (float4 within row)
#pragma unroll
    for (int iter = 0; iter < 8; ++iter) {
        int e  = iter * 256 + tid;
        int lr = e >> 5;
        int q  = e & 31;
        int j  = j0 + lr;
        if (j < KSEL) {
            int row = idx[j];
            row = row < 0 ? 0 : (row >= n ? n - 1 : row);   // safety clamp
            const float* gsrc = embs + (size_t)row * NFEATS + 4 * q;
            float* ldst = &tile[lr * TSTRIDE + 4 * q];
#ifdef HAVE_ASYNC_LDS
            __builtin_amdgcn_global_load_async_to_lds_b128(
                (gp_v4i)gsrc, (lp_v4i)ldst, 0, 0);
#else
            *(float4*)ldst = *(const float4*)gsrc;
#endif
        }
    }
#ifdef HAVE_ASYNC_LDS
    wait_asynccnt0();                      // own wave's async copies done
#endif
    __syncthreads();                       // all waves' copies visible

    // Phase 2: transposed, coalesced-in-j store with tanh gating.
    const int jl = tid & 63;
    const int fb = tid >> 6;               // 0..3
    const int j  = j0 + jl;
    if (j < KSEL) {
        const float g = tanhf(vals[j]);
        for (int f = fb; f < NFEATS; f += 4)
            out[(size_t)f * KSEL + j] = tile[jl * TSTRIDE + f] * g;
    }
}

// ---------------------------------------------------------------------------
extern "C" void kernel_launch(void* const* d_in, const int* in_sizes, int n_in,
                              void* d_out, int out_size, void* d_ws, size_t ws_size,
                              hipStream_t stream) {
    const float* embs = (const float*)d_in[0];
    const float* w    = (const float*)d_in[1];
    float* out        = (float*)d_out;

    const int n = in_sizes[0] / NFEATS;    // 1,000,000

    // workspace layout (base is 256B-aligned from hipMalloc)
    char*     ws     = (char*)d_ws;
    float*    scores = (float*)ws;                               // n floats
    size_t    off    = (size_t)n * sizeof(float);                // 4,000,000
    unsigned* hist   = (unsigned*)(ws + off);        off += NBINS * 4;
    unsigned* meta   = (unsigned*)(ws + off);        off += 256;
    unsigned* bc_gt  = (unsigned*)(ws + off);        off += 256 * 4;
    unsigned* bc_eq  = (unsigned*)(ws + off);        off += 256 * 4;
    int*      idxb   = (int*)(ws + off);             off += (size_t)KSEL * 4;
    float*    valb   = (float*)(ws + off);

    const int nbSel = (n + CHUNK - 1) / CHUNK;       // 123

    topk_init<<<(KSEL + 255) / 256, 256, 0, stream>>>(hist, meta, idxb, valb);
    topk_scores_wmma<<<(n + 127) / 128, 256, 0, stream>>>(embs, w, scores, n);
    topk_hist<<<960, 256, 0, stream>>>(scores, hist, n);
    topk_thresh<<<1, 32, 0, stream>>>(hist, meta);
    topk_count<<<nbSel, 256, 0, stream>>>(scores, meta, bc_gt, bc_eq, n);
    topk_scan<<<1, 32, 0, stream>>>(bc_gt, bc_eq, meta, nbSel);
    topk_write<<<nbSel, 256, 0, stream>>>(scores, meta, bc_gt, bc_eq,
                                          idxb, valb, n);
    topk_gather<<<(KSEL + 63) / 64, 256, 0, stream>>>(embs, idxb, valb, out, n);
}